// SparseMHA_17858474017156
// MI455X (gfx1250) — compile-verified
//
#include <hip/hip_runtime.h>
#include <math.h>

typedef __attribute__((ext_vector_type(2))) float v2f;
typedef __attribute__((ext_vector_type(8))) float v8f;

#define HID   256
#define HEADS 8
#define HD    32
#define RPB   32    // rows per block (2 row-tiles of 16)
#define LDSTR 260   // LDS row stride (floats); 260 % 64 = 4 -> conflict-free A reads

// ---------------------------------------------------------------------------
// [nrows x 256] @ [256 x 256] + bias, scaled, via V_WMMA_F32_16X16X4_F32.
// Block: 256 threads (8 wave32) on a 32-row strip. Each wave owns a 2x2 grid
// of 16x16 tiles (2 col-tiles x 2 row-tiles): per k-step, 2 A + 2 B fragments
// feed 4 WMMAs (register blocking doubles WMMA density per byte loaded).
// ---------------------------------------------------------------------------
__global__ __launch_bounds__(256) void gemm256_wmma(
    const float* __restrict__ X, const float* __restrict__ W,
    const float* __restrict__ B, float* __restrict__ Y,
    float scale, int nrows) {
  __shared__ float a_lds[RPB * LDSTR];

  const int row0 = blockIdx.x * RPB;
  const int t    = threadIdx.x;

  // Cooperative, coalesced load of the 32x256 X strip into LDS (zero-fill tail).
  for (int i = t; i < RPB * HID; i += 256) {
    const int r = i >> 8;         // 0..31
    const int c = i & 255;        // 0..255
    float val = 0.0f;
    if (row0 + r < nrows) val = X[(size_t)(row0 + r) * HID + c];
    a_lds[r * LDSTR + c] = val;
  }
  __syncthreads();

  const int wave  = t >> 5;           // 0..7
  const int lane  = t & 31;
  const int lhalf = lane & 15;        // lane within half-wave
  const int khi   = (lane >> 4) * 2;  // 0 for lanes 0-15, 2 for lanes 16-31
  const int col0  = wave * 32;        // this wave covers cols [col0, col0+32)

  v8f acc[2][2] = {};                 // [col-tile][row-tile]
  const float* wcol0 = W + col0 + lhalf;        // col tile 0
  const float* wcol1 = W + col0 + 16 + lhalf;   // col tile 1

  #pragma unroll 2
  for (int k = 0; k < HID; k += 4) {
    // A fragments (16x4): lane<16 -> K=k,k+1 of row M=lhalf; lane>=16 -> K=k+2,k+3.
    v2f a0, a1, b0, b1;
    a0.x = a_lds[lhalf * LDSTR + k + khi];
    a0.y = a_lds[lhalf * LDSTR + k + khi + 1];
    a1.x = a_lds[(16 + lhalf) * LDSTR + k + khi];
    a1.y = a_lds[(16 + lhalf) * LDSTR + k + khi + 1];
    // B fragments (4x16): lane<16 -> rows K=k,k+1 at col N=lhalf; lane>=16 -> K=k+2,k+3.
    b0.x = wcol0[(size_t)(k + khi) * HID];
    b0.y = wcol0[(size_t)(k + khi + 1) * HID];
    b1.x = wcol1[(size_t)(k + khi) * HID];
    b1.y = wcol1[(size_t)(k + khi + 1) * HID];

    acc[0][0] = __builtin_amdgcn_wmma_f32_16x16x4_f32(false, a0, false, b0,
                                                      (short)0, acc[0][0], false, false);
    acc[0][1] = __builtin_amdgcn_wmma_f32_16x16x4_f32(false, a1, false, b0,
                                                      (short)0, acc[0][1], false, false);
    acc[1][0] = __builtin_amdgcn_wmma_f32_16x16x4_f32(false, a0, false, b1,
                                                      (short)0, acc[1][0], false, false);
    acc[1][1] = __builtin_amdgcn_wmma_f32_16x16x4_f32(false, a1, false, b1,
                                                      (short)0, acc[1][1], false, false);
  }

  // C/D layout: VGPR g -> M = g (lanes 0-15) or M = 8+g (lanes 16-31), N = lhalf.
  #pragma unroll
  for (int ct = 0; ct < 2; ++ct) {
    const int cc   = col0 + ct * 16 + lhalf;
    const float bias = B[cc];
    #pragma unroll
    for (int rt = 0; rt < 2; ++rt) {
      const int rbase = row0 + rt * 16 + ((lane >> 4) ? 8 : 0);
      #pragma unroll
      for (int g = 0; g < 8; ++g) {
        const int r = rbase + g;
        if (r < nrows)
          Y[(size_t)r * HID + cc] = (acc[ct][rt][g] + bias) * scale;
      }
    }
  }
}

// ---------------------------------------------------------------------------
// init: m = -inf, z = 0, acc = 0
// ---------------------------------------------------------------------------
__global__ void init_buffers(float* __restrict__ m, float* __restrict__ z,
                             float* __restrict__ acc, int n) {
  const int t = blockIdx.x * blockDim.x + threadIdx.x;
  if (t < n * HEADS) { m[t] = -INFINITY; z[t] = 0.0f; }
  if (t < n * HID)   { acc[t] = 0.0f; }
}

// ---------------------------------------------------------------------------
// SDDMM: s[e,h] = sum_d q[rows[e], d*8+h] * k[cols[e], d*8+h]; segment max.
// One thread per (edge, head).
// ---------------------------------------------------------------------------
__global__ void sddmm_kernel(const float* __restrict__ q, const float* __restrict__ k,
                             const int* __restrict__ rows, const int* __restrict__ cols,
                             float* __restrict__ s, float* __restrict__ m, int nE) {
  const int t = blockIdx.x * blockDim.x + threadIdx.x;
  if (t >= nE * HEADS) return;
  const int e = t >> 3, h = t & 7;
  const int r = rows[e], c = cols[e];
  const float* qp = q + (size_t)r * HID + h;
  const float* kp = k + (size_t)c * HID + h;
  float acc = 0.0f;
  #pragma unroll
  for (int d = 0; d < HD; ++d) acc = fmaf(qp[d * HEADS], kp[d * HEADS], acc);
  s[t] = acc;
  atomicMax(&m[r * HEADS + h], acc);   // global_atomic_max_num_f32
}

// ---------------------------------------------------------------------------
// exp(s - m[row]) in place; segment sum into z.
// ---------------------------------------------------------------------------
__global__ void edge_exp_kernel(float* __restrict__ s, const float* __restrict__ m,
                                float* __restrict__ z, const int* __restrict__ rows,
                                int nE) {
  const int t = blockIdx.x * blockDim.x + threadIdx.x;
  if (t >= nE * HEADS) return;
  const int e = t >> 3, h = t & 7;
  const int r = rows[e];
  const float val = expf(s[t] - m[r * HEADS + h]);
  s[t] = val;
  atomicAdd(&z[r * HEADS + h], val);
}

// ---------------------------------------------------------------------------
// SPMM: acc[rows[e], c] += (attn[e,h]/z[rows[e],h]) * v[cols[e], c], h = c&7.
// One block (256 threads) per edge: coalesced v reads and acc atomics.
// ---------------------------------------------------------------------------
__global__ __launch_bounds__(256) void spmm_kernel(
    const float* __restrict__ attn, const float* __restrict__ z,
    const float* __restrict__ v, const int* __restrict__ rows,
    const int* __restrict__ cols, float* __restrict__ acc) {
  const int e = blockIdx.x;
  const int c = threadIdx.x;
  const int r = rows[e], src = cols[e];
  const int h = c & 7;
  const float w = attn[(size_t)e * HEADS + h] / z[r * HEADS + h];
  atomicAdd(&acc[(size_t)r * HID + c], w * v[(size_t)src * HID + c]);
}

// ---------------------------------------------------------------------------
extern "C" void kernel_launch(void* const* d_in, const int* in_sizes, int n_in,
                              void* d_out, int out_size, void* d_ws, size_t ws_size,
                              hipStream_t stream) {
  const float* h  = (const float*)d_in[0];
  const float* Wq = (const float*)d_in[1];
  const float* bq = (const float*)d_in[2];
  const float* Wk = (const float*)d_in[3];
  const float* bk = (const float*)d_in[4];
  const float* Wv = (const float*)d_in[5];
  const float* bv = (const float*)d_in[6];
  const float* Wo = (const float*)d_in[7];
  const float* bo = (const float*)d_in[8];
  const int* rows = (const int*)d_in[9];
  const int* cols = (const int*)d_in[10];
  float* out = (float*)d_out;

  const int N = in_sizes[0] / HID;
  const int E = in_sizes[9];
  const float scaling = 1.0f / sqrtf((float)HD);

  // Workspace layout (fp32)
  float* ws  = (float*)d_ws;
  float* q   = ws;                      // N*256
  float* k   = q + (size_t)N * HID;     // N*256
  float* v   = k + (size_t)N * HID;     // N*256
  float* s   = v + (size_t)N * HID;     // E*8 (scores, then attn in place)
  float* m   = s + (size_t)E * HEADS;   // N*8
  float* z   = m + (size_t)N * HEADS;   // N*8
  float* acc = z + (size_t)N * HEADS;   // N*256

  const int mblocks = (N + RPB - 1) / RPB;

  init_buffers<<<(N * HID + 255) / 256, 256, 0, stream>>>(m, z, acc, N);

  gemm256_wmma<<<mblocks, 256, 0, stream>>>(h, Wq, bq, q, scaling, N);
  gemm256_wmma<<<mblocks, 256, 0, stream>>>(h, Wk, bk, k, 1.0f, N);
  gemm256_wmma<<<mblocks, 256, 0, stream>>>(h, Wv, bv, v, 1.0f, N);

  const int et = (E * HEADS + 255) / 256;
  sddmm_kernel<<<et, 256, 0, stream>>>(q, k, rows, cols, s, m, E);
  edge_exp_kernel<<<et, 256, 0, stream>>>(s, m, z, rows, E);
  spmm_kernel<<<E, 256, 0, stream>>>(s, z, v, rows, cols, acc);

  gemm256_wmma<<<mblocks, 256, 0, stream>>>(acc, Wo, bo, out, 1.0f, N);
}